// MeanPool_54133767798855
// MI455X (gfx1250) — compile-verified
//
#include <hip/hip_runtime.h>
#include <hip/hip_bf16.h>

typedef __attribute__((ext_vector_type(2))) float v2f;
typedef __attribute__((ext_vector_type(4))) float v4f;
typedef __attribute__((ext_vector_type(8))) float v8f;

#define B_IMGS   256
#define D_DIM    512
#define HW       256           // H*W
#define T_TOT    32768         // B*SEG_LEN
#define HALF_F4  16777216u     // (256*256*512)/2 float4: base offset of the snd output half

// ---------------------------------------------------------------------------
// Kernel A: prefix-sum of snd_splits -> segment offsets + lengths (tiny)
// ---------------------------------------------------------------------------
__global__ void seg_offsets_kernel(const int* __restrict__ splits,
                                   int* __restrict__ segoff,
                                   int* __restrict__ seglen) {
    int t = threadIdx.x;
    if (t < B_IMGS) {
        int off = 0;
        for (int u = 0; u < t; ++u) off += splits[u];
        segoff[t] = off;
        seglen[t] = splits[t];
    }
}

// ---------------------------------------------------------------------------
// Kernel B: spatial mean pool via V_WMMA_F32_16X16X4_F32 (ones-matmul).
// Each wave reduces 16 flattened rows (row = b*512+d, row length 256).
// A layout (16x4 f32): lane l -> M = l&15, K = 2*(l>>4)+{0,1}  (float2 load)
// B = ones => D[m,n] = rowsum(m) for all n.
// ---------------------------------------------------------------------------
__global__ void img_meanpool_wmma(const float* __restrict__ Zimg,
                                  float* __restrict__ zimg) {
    const int lane = threadIdx.x & 31;
    const int wave = blockIdx.x * (blockDim.x >> 5) + (threadIdx.x >> 5);
    const int r0   = wave << 4;                    // 16 rows per wave
    const int m    = lane & 15;
    const int koff = (lane >> 4) << 1;             // 0 or 2
    const float* rowp = Zimg + (size_t)(r0 + m) * HW + koff;

    v8f c = {};
    v2f ones; ones.x = 1.0f; ones.y = 1.0f;
    #pragma unroll 4
    for (int k = 0; k < HW; k += 4) {
        v2f a = *(const v2f*)(rowp + k);           // 8B aligned
        c = __builtin_amdgcn_wmma_f32_16x16x4_f32(
                false, a, false, ones, (short)0, c, false, false);
    }
    // c[i]: lane 0 -> M=i, lane 16 -> M=8+i (all N identical)
    if ((lane & 15) == 0) {
        const int mb = (lane >> 4) << 3;
        #pragma unroll
        for (int i = 0; i < 8; ++i)
            zimg[r0 + mb + i] = c[i] * (1.0f / (float)HW);
    }
}

// ---------------------------------------------------------------------------
// Kernel C: ragged segment mean via the same ones-matmul.
// A[m,k] = Zsnd[off+k, c0+m]  => D[m,*] = column sum over segment rows.
// Tail rows predicated with selects so EXEC stays all-1s for WMMA.
// Wave handles one (segment, 16-column group). 256 segs * 32 groups = 8192 waves.
// ---------------------------------------------------------------------------
__global__ void snd_segmean_wmma(const float* __restrict__ Zsnd,
                                 const int* __restrict__ segoff,
                                 const int* __restrict__ seglen,
                                 float* __restrict__ zsnd) {
    const int lane = threadIdx.x & 31;
    const int wave = blockIdx.x * (blockDim.x >> 5) + (threadIdx.x >> 5);
    const int seg  = wave >> 5;
    const int c0   = (wave & 31) << 4;
    const int off  = segoff[seg];
    const int len  = seglen[seg];
    const int m    = lane & 15;
    const int k0   = (lane >> 4) << 1;             // 0 or 2
    const float* base = Zsnd + (size_t)off * D_DIM + (c0 + m);

    v8f c = {};
    v2f ones; ones.x = 1.0f; ones.y = 1.0f;
    for (int k = 0; k < len; k += 4) {
        const int ka = k + k0;
        const int kb = ka + 1;
        v2f a;
        a.x = (ka < len) ? base[(size_t)ka * D_DIM] : 0.0f;
        a.y = (kb < len) ? base[(size_t)kb * D_DIM] : 0.0f;
        c = __builtin_amdgcn_wmma_f32_16x16x4_f32(
                false, a, false, ones, (short)0, c, false, false);
    }
    if ((lane & 15) == 0) {
        const float inv = 1.0f / (float)len;
        const int mb = (lane >> 4) << 3;
        #pragma unroll
        for (int i = 0; i < 8; ++i)
            zsnd[(size_t)seg * D_DIM + c0 + mb + i] = c[i] * inv;
    }
}

// ---------------------------------------------------------------------------
// Kernel D: broadcast writes (the 256 MiB bottleneck).
// M_img[i,j,d] = z_img[j,d] ; M_snd[i,j,d] = z_snd[i,d].
// Thread loads one float4 from the L2-resident pooled array and replicates it
// 4x along the broadcast dim with non-temporal B128 stores.
// Blocks [0,8192): img half.  [8192,16384): snd half.
// ---------------------------------------------------------------------------
__global__ void broadcast_out_kernel(const float* __restrict__ zimg,
                                     const float* __restrict__ zsnd,
                                     v4f* __restrict__ out) {
    const bool sndHalf = (blockIdx.x >= 8192);
    const unsigned tid = (sndHalf ? (blockIdx.x - 8192) : blockIdx.x) * blockDim.x
                         + threadIdx.x;            // [0, 2097152)
    const unsigned rb  = tid >> 15;                // replicate block 0..63
    const unsigned xd  = tid & 32767u;
    const unsigned x   = xd >> 7;                  // j (img half) or i (snd half)
    const unsigned d4  = xd & 127u;                // float4 index within a 512-f row

    const v4f src = sndHalf ? ((const v4f*)zsnd)[x * 128u + d4]
                            : ((const v4f*)zimg)[x * 128u + d4];

    // output float4 index: i*32768 + j*128 + d4 (+ half offset)
    size_t o, stride;
    if (sndHalf) { o = (size_t)HALF_F4 + (size_t)x * 32768u + d4; stride = 128u; }
    else         { o = (size_t)x * 128u + d4;                     stride = 32768u; }
    o += (size_t)(rb << 2) * stride;

    #pragma unroll
    for (int t = 0; t < 4; ++t) {
        __builtin_nontemporal_store(src, &out[o]);
        o += stride;
    }
}

// ---------------------------------------------------------------------------
extern "C" void kernel_launch(void* const* d_in, const int* in_sizes, int n_in,
                              void* d_out, int out_size, void* d_ws, size_t ws_size,
                              hipStream_t stream) {
    const float* Zimg   = (const float*)d_in[0];   // [256, 512, 16, 16] f32
    const float* Zsnd   = (const float*)d_in[1];   // [32768, 512] f32
    const int*   splits = (const int*)d_in[2];     // [256] int

    // workspace layout
    float* zimg = (float*)d_ws;                                   // 131072 f
    float* zsnd = zimg + (size_t)B_IMGS * D_DIM;                  // 131072 f
    int*   segoff = (int*)(zsnd + (size_t)B_IMGS * D_DIM);        // 256 i
    int*   seglen = segoff + B_IMGS;                              // 256 i

    seg_offsets_kernel<<<1, 256, 0, stream>>>(splits, segoff, seglen);

    // 131072 rows / 16 per wave = 8192 waves; 4 waves (128 thr) per block
    img_meanpool_wmma<<<2048, 128, 0, stream>>>(Zimg, zimg);

    // 256 segments * 32 column-groups = 8192 waves
    snd_segmean_wmma<<<2048, 128, 0, stream>>>(Zsnd, segoff, seglen, zsnd);

    // 2 halves * 2,097,152 threads, each writing 4 float4 (16 floats)
    broadcast_out_kernel<<<16384, 256, 0, stream>>>(zimg, zsnd, (v4f*)d_out);
}